// PairwiseLoss_69956427317340
// MI455X (gfx1250) — compile-verified
//
#include <hip/hip_runtime.h>
#include <stdint.h>

// ---------------------------------------------------------------------------
// PairwiseLoss for MI455X (gfx1250):
//   134 MB fp32 streaming reduction, ~5.8 us HBM floor @ 23.3 TB/s, balanced
//   against ~235M VALU lane-ops (2 transcendentals/pair).  Data is streamed
//   global -> LDS with the CDNA5 async-DMA path (ASYNCcnt, depth-3 pipeline),
//   labels come from an LDS-resident gallery table, per-wave reduction uses
//   wave32 shuffles, and the cross-wave block reduction uses
//   V_WMMA_F32_16X16X4_F32 with a ones A-matrix (D[m,n] = sum_k B[k,n]).
// ---------------------------------------------------------------------------

typedef __attribute__((ext_vector_type(2))) float v2f;
typedef __attribute__((ext_vector_type(8))) float v8f;

#define TPB 256
#define NBLK 1024
#define GSZ 4096
#define NPAIRS (4096u * 4096u)
#define PPB (NPAIRS / NBLK)                  /* 16384 pairs per block      */
#define PAIRS_PER_STAGE (TPB * 2)            /* 512 pairs / 4 KB per stage */
#define NSTAGES (PPB / PAIRS_PER_STAGE)      /* 32                          */
#define ROWS_PER_BLK (PPB / GSZ)             /* 4 probe rows per block      */
#define STAGES_PER_ROW (NSTAGES / ROWS_PER_BLK) /* 8                        */

// Generic pointers into LDS carry the wave-relative LDS byte offset in their
// low 32 bits, which is exactly what
// the async-DMA VDST VGPR wants.
__device__ __forceinline__ uint32_t lds_lo32(const void* p) {
  return (uint32_t)(uintptr_t)p;
}

__device__ __forceinline__ void async_g2l_b128(uint32_t ldsoff, const float4* g) {
  asm volatile("global_load_async_to_lds_b128 %0, %1, off"
               :
               : "v"(ldsoff), "v"(g)
               : "memory");
}

template <int N>
__device__ __forceinline__ void s_wait_async() {
  asm volatile("s_wait_asynccnt %0" ::"i"(N) : "memory");
}
__device__ __forceinline__ void s_wait_ds0() {
  asm volatile("s_wait_dscnt 0" ::: "memory");
}

__device__ __forceinline__ float fast_exp2(float x) {
#if __has_builtin(__builtin_amdgcn_exp2f)
  return __builtin_amdgcn_exp2f(x);
#else
  return exp2f(x);
#endif
}
__device__ __forceinline__ float fast_log2(float x) {
#if __has_builtin(__builtin_amdgcn_logf)
  return __builtin_amdgcn_logf(x);   // v_log_f32 computes log2
#else
  return __log2f(x);
#endif
}

__device__ __forceinline__ void do_pair(float l0, float l1, bool pos,
                                        float& apos, float& aneg,
                                        int& npos, int& ncor) {
  const float LOG2E = 1.4426950408889634f;
  const float LN2 = 0.6931471805599453f;
  float m = fmaxf(l0, l1);
  float d = fminf(l0, l1) - m;                 // <= 0
  float e = fast_exp2(d * LOG2E);              // exp(d)
  float u = fast_log2(1.0f + e);               // log2(1+e)
  float lse = fmaf(u, LN2, m);                 // logsumexp(l0,l1)
  float nll = lse - (pos ? l1 : l0);
  apos += pos ? nll : 0.0f;
  aneg += pos ? 0.0f : nll;
  npos += pos ? 1 : 0;
  bool pred1 = l1 > l0;                        // argmax ties -> class 0
  ncor += (pred1 == pos) ? 1 : 0;
}

__global__ __launch_bounds__(TPB) void pl_phase1(const float* __restrict__ cls,
                                                 const int* __restrict__ tprobe,
                                                 const int* __restrict__ tgal,
                                                 float* __restrict__ part) {
  __shared__ float4 abuf[4][TPB];   // 16 KB async staging ring
  __shared__ int gal[GSZ];          // 16 KB gallery labels
  __shared__ float red[8][4];       // per-wave partials

  const int t = threadIdx.x;
  const int b = blockIdx.x;

  for (int k = t; k < GSZ; k += TPB) gal[k] = tgal[k];
  __syncthreads();

  const float4* src = (const float4*)cls + (size_t)b * (PPB / 2);
  const uint32_t lbase = lds_lo32(&abuf[0][t]);
  const uint32_t lstep = (uint32_t)(TPB * sizeof(float4));

  // Prologue: 3 stages in flight on the async engine.
#pragma unroll
  for (int s = 0; s < 3; ++s)
    async_g2l_b128(lbase + (uint32_t)s * lstep, src + s * TPB + t);

  float apos = 0.0f, aneg = 0.0f;
  int npos = 0, ncor = 0;

  for (int r = 0; r < ROWS_PER_BLK; ++r) {
    const int P = tprobe[b * ROWS_PER_BLK + r];  // uniform -> SGPR
    for (int si = 0; si < STAGES_PER_ROW; ++si) {
      const int s = r * STAGES_PER_ROW + si;
      const int rem = NSTAGES - 1 - s;           // stages still behind this one
      if (rem >= 2)      s_wait_async<2>();      // stage s complete
      else if (rem == 1) s_wait_async<1>();
      else               s_wait_async<0>();

      float4 v = abuf[s & 3][t];                 // ds_load_b128 (thread-private slot)
      s_wait_ds0();                              // data in VGPRs before buffer reuse

      const int nxt = s + 3;
      if (nxt < NSTAGES)
        async_g2l_b128(lbase + (uint32_t)(nxt & 3) * lstep, src + nxt * TPB + t);

      const int j = (si * PAIRS_PER_STAGE + 2 * t) & (GSZ - 1);
      const int2 g = *(const int2*)&gal[j];
      do_pair(v.x, v.y, g.x == P, apos, aneg, npos, ncor);
      do_pair(v.z, v.w, g.y == P, apos, aneg, npos, ncor);
    }
  }

  // wave32 reduction of the 4 accumulators.
  float q0 = apos, q1 = aneg, q2 = (float)npos, q3 = (float)ncor;
#pragma unroll
  for (int off = 16; off > 0; off >>= 1) {
    q0 += __shfl_xor(q0, off, 32);
    q1 += __shfl_xor(q1, off, 32);
    q2 += __shfl_xor(q2, off, 32);
    q3 += __shfl_xor(q3, off, 32);
  }
  const int lane = t & 31, wave = t >> 5;
  if (lane == 0) {
    red[wave][0] = q0; red[wave][1] = q1; red[wave][2] = q2; red[wave][3] = q3;
  }
  __syncthreads();

  if (t < 32) {  // wave 0 only: EXEC all-ones within the wave, as WMMA requires
    // Cross-wave reduce via V_WMMA_F32_16X16X4_F32, A = ones:
    //   D[m][n] = sum_k B[k][n], so sum_n D[0][n] == sum of every B element —
    //   independent of the exact B lane layout.  Row 0 of D sits in d[0] of
    //   lanes 0..15 (documented C/D tile layout), so 4 xor-shuffles finish it.
    v2f a; a[0] = 1.0f; a[1] = 1.0f;
    float tot[4];
#pragma unroll
    for (int q = 0; q < 4; ++q) {
      v2f bm; bm[0] = (t < 8) ? red[t][q] : 0.0f; bm[1] = 0.0f;
      v8f c = {};
      v8f d = __builtin_amdgcn_wmma_f32_16x16x4_f32(false, a, false, bm,
                                                    (short)0, c, false, false);
      float x = d[0];
      x += __shfl_xor(x, 1, 32);
      x += __shfl_xor(x, 2, 32);
      x += __shfl_xor(x, 4, 32);
      x += __shfl_xor(x, 8, 32);
      tot[q] = x;
    }
    if (t == 0) {
      *(float4*)&part[(size_t)b * 4] = make_float4(tot[0], tot[1], tot[2], tot[3]);
    }
  }
}

__global__ __launch_bounds__(256) void pl_phase2(const float* __restrict__ part,
                                                 float* __restrict__ out) {
  __shared__ float s4[256];
  __shared__ double dq[4];
  const int t = threadIdx.x;
  const int q = t & 3;
  float acc = 0.0f;
  for (int i = t >> 2; i < NBLK; i += 64) acc += part[i * 4 + q];
  s4[t] = acc;
  __syncthreads();
  if (t < 4) {
    double d = 0.0;
    for (int k = t; k < 256; k += 4) d += (double)s4[k];
    dq[t] = d;
  }
  __syncthreads();
  if (t == 0) {
    const double W_NEG = 1.0 / 4095.0;  // SAMPLING_RATE / (BATCHSIZE - 1)
    const double N = (double)NPAIRS;
    const double posnll = dq[0], negnll = dq[1], np = dq[2], nc = dq[3];
    const double sum_w = np + (N - np) * W_NEG;
    out[0] = (float)((posnll + W_NEG * negnll) / sum_w);  // loss
    out[1] = (float)(nc / N * 100.0);                     // top-1 accuracy %
  }
}

extern "C" void kernel_launch(void* const* d_in, const int* in_sizes, int n_in,
                              void* d_out, int out_size, void* d_ws, size_t ws_size,
                              hipStream_t stream) {
  (void)in_sizes; (void)n_in; (void)out_size; (void)ws_size;
  const float* cls = (const float*)d_in[0];
  const int* tprobe = (const int*)d_in[1];
  const int* tgal = (const int*)d_in[2];
  float* out = (float*)d_out;
  float* part = (float*)d_ws;  // NBLK * 4 floats = 16 KB scratch

  pl_phase1<<<NBLK, TPB, 0, stream>>>(cls, tprobe, tgal, part);
  pl_phase2<<<1, 256, 0, stream>>>(part, out);
}